// ParallelEnergyNet_81827716923448
// MI455X (gfx1250) — compile-verified
//
#include <hip/hip_runtime.h>
#include <hip/hip_bf16.h>
#include <stdint.h>

#define NN     8        // neighbors per center
#define NSPEC  8        // species rows in embed table
#define NP     56       // NN*(NN-1) off-diagonal pairs
#define FEAT   64       // embedding features
#define DDESC  195      // 3 + 3*FEAT
#define BLOCK  256      // 8 wave32 per workgroup

typedef float v2f __attribute__((ext_vector_type(2)));
typedef float v4f __attribute__((ext_vector_type(4)));
typedef float v8f __attribute__((ext_vector_type(8)));

__global__ __launch_bounds__(BLOCK)
void ang_desc_kernel(const int*   __restrict__ atom_i_idx,
                     const int*   __restrict__ atom_j_idx,
                     const float* __restrict__ dist_ij,
                     const float* __restrict__ atoms_xyz,
                     const int*   __restrict__ atoms_long,
                     const float* __restrict__ embed_table,
                     float*       __restrict__ out,
                     int nC)
{
    __shared__ __align__(16) float sBlock[NP * DDESC]; // 43,680 B staging tile
    __shared__ __align__(16) float sE[NSPEC * FEAT];   // embedding table (2 KB)
    __shared__ float sdist[NN];
    __shared__ float srinv[NN];
    __shared__ float sx[NN], sy[NN], sz[NN];
    __shared__ int   sspec[NN];
    __shared__ int   sspec_i;
    __shared__ float stjk[NN * NN];

    const int c   = blockIdx.x;
    const int tid = threadIdx.x;
    if (c >= nC) return;                       // block-uniform

    // ---- CDNA5 async path: DMA the 2KB embed table into LDS (ASYNCcnt) ----
    if (tid < (NSPEC * FEAT * 4) / 16) {       // 128 lanes x 16B
        unsigned    ldsoff = (unsigned)(uintptr_t)(&sE[0]) + (unsigned)tid * 16u;
        const char* gsrc   = (const char*)embed_table + (size_t)tid * 16;
        asm volatile("global_load_async_to_lds_b128 %0, %1, off"
                     :: "v"(ldsoff), "v"(gsrc) : "memory");
    }

    // ---- per-center tiny gathers (overlap with the async DMA) ----
    if (tid < NN) {
        float d    = dist_ij[(size_t)c * NN + tid];
        sdist[tid] = d;
        srinv[tid] = 1.0f / d;                 // IEEE rcp, hoisted once
        int aj     = atom_j_idx[(size_t)c * NN + tid];
        sspec[tid] = atoms_long[(size_t)aj * 2 + 1];
        sx[tid]    = atoms_xyz[(size_t)aj * 3 + 0];
        sy[tid]    = atoms_xyz[(size_t)aj * 3 + 1];
        sz[tid]    = atoms_xyz[(size_t)aj * 3 + 2];
    } else if (tid == NN) {
        int ai  = atom_i_idx[c];
        sspec_i = atoms_long[(size_t)ai * 2 + 1];
        out[c]  = (float)ai;                   // output 0: ids as float
    }

    asm volatile("s_wait_asynccnt 0" ::: "memory");
    __syncthreads();

    // ---- t_jk table (needs dist/xyz) ----
    if (tid < NN * NN) {
        int   j  = tid >> 3, k = tid & 7;
        float dx = sx[j] - sx[k], dy = sy[j] - sy[k], dz = sz[j] - sz[k];
        float djk = sqrtf(dx * dx + dy * dy + dz * dz);
        float dj = sdist[j], dk = sdist[k];
        float mx = fmaxf(dj, dk), mn = fminf(dj, dk);
        stjk[tid] = (djk - mx + mn) / (2.0f * mn);
    }
    __syncthreads();

    // ---- scalar columns [t_ij, t_ik, t_jk] of the 56x195 block ----
    if (tid < NP * 3) {
        int p = tid / 3, f = tid - 3 * p;
        int j = p / (NN - 1);
        int r = p - j * (NN - 1);
        int k = r + (r >= j ? 1 : 0);
        float val = (f == 0) ? sdist[j] : (f == 1) ? sdist[k] : stjk[j * NN + k];
        sBlock[p * DDESC + f] = val;
    }

    // ---- embedding blocks via WMMA: D(64x64) = R(64x8) @ E(8x64) ----
    // q=0: R = onehot(spec_i); q=1: R[p,s]=(s==spec_j)?1/t_ij:0; q=2: same w/ k.
    // 3 matrices x (4x4) 16x16 tiles = 48 jobs, exactly 6 per wave (8 waves):
    // every wave runs the same trip count, EXEC stays all-ones for WMMA.
    {
        const int wave    = tid >> 5;
        const int lane    = tid & 31;
        const int halfsel = lane >> 4;   // 0: lanes 0-15, 1: lanes 16-31
        const int mn      = lane & 15;   // A-row (M) / B,C-col (N)

        for (int job = wave; job < 48; job += 8) {
            const int q  = job >> 4;            // which embedding block
            const int t  = job & 15;
            const int m0 = (t >> 2) << 4;       // pair-row tile base
            const int n0 = (t & 3) << 4;        // feature-col tile base

            // this lane's A-row -> pair index (masked past 56)
            int  p     = m0 + mn;
            bool valid = p < NP;
            int  pp    = valid ? p : 0;
            int  j     = pp / (NN - 1);
            int  r     = pp - j * (NN - 1);
            int  k     = r + (r >= j ? 1 : 0);

            int   sel;
            float scale;
            if (q == 0)      { sel = sspec_i;  scale = 1.0f;     }   // wave-uniform q
            else if (q == 1) { sel = sspec[j]; scale = srinv[j]; }
            else             { sel = sspec[k]; scale = srinv[k]; }
            scale = valid ? scale : 0.0f;

            // A 16x4 f32: lane L -> M=L%16, K=2*(L/16)+v ; two k-steps k0=0,4
            // B 4x16 f32: lane L -> N=L%16, K=2*(L/16)+v
            v2f a0, a1, b0, b1;
#pragma unroll
            for (int v = 0; v < 2; ++v) {
                int K0 = 2 * halfsel + v;       // 0..3
                int K1 = 4 + K0;                // 4..7
                a0[v] = (K0 == sel) ? scale : 0.0f;
                a1[v] = (K1 == sel) ? scale : 0.0f;
                b0[v] = sE[K0 * FEAT + n0 + mn];
                b1[v] = sE[K1 * FEAT + n0 + mn];
            }

            v8f cfrag = {};
            cfrag = __builtin_amdgcn_wmma_f32_16x16x4_f32(
                        false, a0, false, b0, (short)0, cfrag, false, false);
            cfrag = __builtin_amdgcn_wmma_f32_16x16x4_f32(
                        false, a1, false, b1, (short)0, cfrag, false, false);

            // C/D 16x16 f32: VGPR v, lane L -> M = v + 8*(L/16), N = L%16
            const int colbase = 3 + q * FEAT + n0 + mn;
#pragma unroll
            for (int v = 0; v < 8; ++v) {
                int prow = m0 + v + 8 * halfsel;
                if (prow < NP)
                    sBlock[prow * DDESC + colbase] = cfrag[v];
            }
        }
    }
    __syncthreads();

    // ---- stream the 43.7KB center block: coalesced 16B NT stores ----
    // base byte offset = 4*nC + c*43680, both multiples of 16.
    float* outc = out + nC + (size_t)c * (NP * DDESC);
    const v4f* sb4 = (const v4f*)sBlock;
    const int total4 = (NP * DDESC) / 4;        // 2730
    for (int qq = tid; qq < total4; qq += BLOCK) {
        v4f v = sb4[qq];
        // 437MB output > 192MB L2 and never re-read on device: bypass with NT
        __builtin_nontemporal_store(v, (v4f*)outc + qq);
    }
}

extern "C" void kernel_launch(void* const* d_in, const int* in_sizes, int n_in,
                              void* d_out, int out_size, void* d_ws, size_t ws_size,
                              hipStream_t stream) {
    const int*   atom_i = (const int*)  d_in[0];
    const int*   atom_j = (const int*)  d_in[1];
    const float* dist   = (const float*)d_in[2];
    const float* xyz    = (const float*)d_in[3];
    const int*   along  = (const int*)  d_in[4];
    const float* embed  = (const float*)d_in[5];
    // d_in[6] = nNeigh scalar (fixed to 8 by the problem shape)
    const int nC = in_sizes[0];

    ang_desc_kernel<<<nC, BLOCK, 0, stream>>>(atom_i, atom_j, dist, xyz, along,
                                              embed, (float*)d_out, nC);
}